// FGDNBranch_80401787781631
// MI455X (gfx1250) — compile-verified
//
#include <hip/hip_runtime.h>

// ---------------------------------------------------------------------------
// ChebConv(K=3) x2 + PReLU + global mean pool, for MI455X (gfx1250, wave32).
//
// Roofline: ~6.5 GB of edge-scatter traffic dominates (23.3 TB/s HBM, features
// L2-resident) vs only ~15 GFLOP of GEMM -> keep fp32 everywhere and use the
// native fp32 WMMA V_WMMA_F32_16X16X4_F32 for the dense node-feature GEMMs.
// ---------------------------------------------------------------------------

#define N_NODES  100000
#define N_EDGES  1600000
#define IN_C     128
#define HID_C    128
#define OUT_C    64
#define N_GRAPHS 64

typedef __attribute__((ext_vector_type(2))) float v2f;
typedef __attribute__((ext_vector_type(8))) float v8f;

// ----------------------------- degree / weights -----------------------------

__global__ void degree_kernel(const int* __restrict__ ei, float* __restrict__ deg, int E) {
    int e = blockIdx.x * blockDim.x + threadIdx.x;
    if (e >= E) return;
    atomicAdd(&deg[ei[e]], 1.0f);   // row = ei[0][e]
}

__global__ void rsqrt_kernel(float* __restrict__ deg, int n) {
    int i = blockIdx.x * blockDim.x + threadIdx.x;
    if (i >= n) return;
    float d = deg[i];
    deg[i] = (d > 0.0f) ? rsqrtf(fmaxf(d, 1.0f)) : 0.0f;   // in-place: deg -> dis
}

__global__ void edge_w_kernel(const int* __restrict__ ei, const float* __restrict__ dis,
                              float* __restrict__ wE, int E) {
    int e = blockIdx.x * blockDim.x + threadIdx.x;
    if (e >= E) return;
    int row = ei[e];
    int col = ei[E + e];
    wE[e] = -(dis[row] * dis[col]);
}

// ----------------------------- Chebyshev propagate --------------------------
// out[col] += scale * w[e] * feat[row]   (feature dim fixed at 128)
// One wave32 per edge: each lane moves a float4 slice (32*4 = 128 channels).
__global__ void prop_kernel(const float* __restrict__ feat, const int* __restrict__ ei,
                            const float* __restrict__ wE, float* __restrict__ outf,
                            float scale, int E) {
    int gwave = (blockIdx.x * blockDim.x + threadIdx.x) >> 5;
    int lane  = threadIdx.x & 31;
    if (gwave >= E) return;                       // wave-uniform
    int   row = ei[gwave];
    int   col = ei[E + gwave];
    float w   = wE[gwave] * scale;
    const float4 v = *(const float4*)(feat + (size_t)row * 128 + lane * 4);
    float* op = outf + (size_t)col * 128 + lane * 4;
    atomicAdd(op + 0, w * v.x);
    atomicAdd(op + 1, w * v.y);
    atomicAdd(op + 2, w * v.z);
    atomicAdd(op + 3, w * v.w);
}

// Tx2 := -Tx0   (so the following prop with scale=2 yields 2*prop(Tx1) - Tx0)
__global__ void init_neg_kernel(const float* __restrict__ src, float* __restrict__ dst, int n) {
    int i = blockIdx.x * blockDim.x + threadIdx.x;
    if (i >= n) return;
    dst[i] = -src[i];
}

// ----------------------------- fused WMMA GEMM ------------------------------
// out = prelu( A0@W[0] + A1@W[1] + A2@W[2] + b , alpha )
// A*: [M, CIN] fp32, W: [3, CIN, COUT] fp32.  One wave per 16x16 output tile.
// Uses V_WMMA_F32_16X16X4_F32; K loop = 3 * CIN/4 iterations.
// A layout (ISA 7.12.2): lanes 0-15 row M=l16 K={kb,kb+1}; lanes 16-31 K={kb+2,kb+3}.
// C/D layout: VGPR j -> rows j (lanes 0-15) and j+8 (lanes 16-31), N = lane%16.
template <int CIN, int COUT>
__global__ void cheb_gemm_wmma(const float* __restrict__ A0, const float* __restrict__ A1,
                               const float* __restrict__ A2, const float* __restrict__ W,
                               const float* __restrict__ bias, const float* __restrict__ alpha,
                               float* __restrict__ out, int M) {
    const int wave = threadIdx.x >> 5;                 // 4 waves / block
    const int lane = threadIdx.x & 31;
    const int mt   = blockIdx.x * 4 + wave;
    const int m0   = mt * 16;
    if (m0 >= M) return;                               // wave-uniform -> EXEC stays full
    const int n0   = blockIdx.y * 16;
    const int half = lane >> 4;                        // 0: K+0,K+1   1: K+2,K+3
    const int l16  = lane & 15;

    v8f acc = {};
    const float* Aptr[3] = {A0, A1, A2};
#pragma unroll
    for (int tx = 0; tx < 3; ++tx) {
        const float* Arow = Aptr[tx] + (size_t)(m0 + l16) * CIN;
        const float* Wcol = W + (size_t)tx * CIN * COUT + n0 + l16;
#pragma unroll 8
        for (int kb = 0; kb < CIN; kb += 4) {
            const int ka = kb + half * 2;
            v2f a = *(const v2f*)(Arow + ka);                          // K=ka, ka+1
            v2f b;
            b.x = Wcol[(size_t)ka * COUT];                             // K=ka
            b.y = Wcol[(size_t)(ka + 1) * COUT];                       // K=ka+1
            acc = __builtin_amdgcn_wmma_f32_16x16x4_f32(
                false, a, false, b, (short)0, acc, false, false);
        }
    }

    const int   col = n0 + l16;
    const float bv  = bias[col];
    const float av  = alpha[col];
#pragma unroll
    for (int j = 0; j < 8; ++j) {
        const int row = m0 + half * 8 + j;
        float v = acc[j] + bv;
        v = (v >= 0.0f) ? v : av * v;
        out[(size_t)row * COUT + col] = v;
    }
}

// ----------------------------- mean pool ------------------------------------

__global__ void pool_accum_kernel(const float* __restrict__ h, const int* __restrict__ batch,
                                  float* __restrict__ sums, float* __restrict__ cnt, int M) {
    int idx  = blockIdx.x * blockDim.x + threadIdx.x;   // 16 threads per node
    int node = idx >> 4;
    int part = idx & 15;                                // 4 channels each (OUT_C=64)
    if (node >= M) return;
    int g = batch[node];
    if (part == 0) atomicAdd(&cnt[g], 1.0f);
    const float* hp = h + (size_t)node * OUT_C + part * 4;
    float* sp = sums + (size_t)g * OUT_C + part * 4;
#pragma unroll
    for (int i = 0; i < 4; ++i) atomicAdd(&sp[i], hp[i]);
}

__global__ void pool_final_kernel(const float* __restrict__ sums, const float* __restrict__ cnt,
                                  float* __restrict__ out) {
    int i = blockIdx.x * blockDim.x + threadIdx.x;
    if (i >= N_GRAPHS * OUT_C) return;
    int g = i / OUT_C;
    out[i] = sums[i] / fmaxf(cnt[g], 1.0f);
}

// ----------------------------- host launcher --------------------------------

extern "C" void kernel_launch(void* const* d_in, const int* in_sizes, int n_in,
                              void* d_out, int out_size, void* d_ws, size_t ws_size,
                              hipStream_t stream) {
    const float* x     = (const float*)d_in[0];
    const int*   ei    = (const int*)d_in[1];   // [2, E]
    const int*   batch = (const int*)d_in[2];
    const float* W1    = (const float*)d_in[3]; // [3,128,128]
    const float* b1    = (const float*)d_in[4];
    const float* a1    = (const float*)d_in[5];
    const float* W2    = (const float*)d_in[6]; // [3,128,64]
    const float* b2    = (const float*)d_in[7];
    const float* a2    = (const float*)d_in[8];
    float* outp = (float*)d_out;

    (void)in_sizes; (void)n_in; (void)out_size; (void)ws_size;

    // Workspace carve-out (256B aligned).
    size_t off = 0;
    auto carve = [&](size_t bytes) -> float* {
        float* p = (float*)((char*)d_ws + off);
        off += (bytes + 255) & ~(size_t)255;
        return p;
    };
    float* deg     = carve((size_t)N_NODES * 4);           // degrees -> dis (in place)
    float* wE      = carve((size_t)N_EDGES * 4);           // normalized edge weights
    float* Tx1     = carve((size_t)N_NODES * 128 * 4);
    float* Tx2     = carve((size_t)N_NODES * 128 * 4);
    float* hbuf    = carve((size_t)N_NODES * HID_C * 4);   // layer-1 output
    float* nodeOut = carve((size_t)N_NODES * OUT_C * 4);   // layer-2 output
    float* sums    = carve((size_t)N_GRAPHS * OUT_C * 4);
    float* cnt     = carve((size_t)N_GRAPHS * 4);

    const int TPB = 256;
    const int eBlocks    = (N_EDGES + TPB - 1) / TPB;
    const int nBlocks    = (N_NODES + TPB - 1) / TPB;
    const int propBlocks = (N_EDGES * 32 + TPB - 1) / TPB;     // 1 wave per edge
    const int featBlocks = ((size_t)N_NODES * 128 + TPB - 1) / TPB;
    const int mTiles     = (N_NODES + 15) / 16;                // 6250
    const dim3 gemmBlk(128);
    const dim3 gemmGrd1((mTiles + 3) / 4, HID_C / 16);
    const dim3 gemmGrd2((mTiles + 3) / 4, OUT_C / 16);

    // --- normalization: deg -> dis -> w ---
    hipMemsetAsync(deg, 0, (size_t)N_NODES * 4, stream);
    degree_kernel<<<eBlocks, TPB, 0, stream>>>(ei, deg, N_EDGES);
    rsqrt_kernel<<<nBlocks, TPB, 0, stream>>>(deg, N_NODES);
    edge_w_kernel<<<eBlocks, TPB, 0, stream>>>(ei, deg, wE, N_EDGES);

    // --- layer 1: Tx0 = x ---
    hipMemsetAsync(Tx1, 0, (size_t)N_NODES * 128 * 4, stream);
    prop_kernel<<<propBlocks, TPB, 0, stream>>>(x, ei, wE, Tx1, 1.0f, N_EDGES);
    init_neg_kernel<<<featBlocks, TPB, 0, stream>>>(x, Tx2, N_NODES * 128);
    prop_kernel<<<propBlocks, TPB, 0, stream>>>(Tx1, ei, wE, Tx2, 2.0f, N_EDGES);
    cheb_gemm_wmma<IN_C, HID_C><<<gemmGrd1, gemmBlk, 0, stream>>>(
        x, Tx1, Tx2, W1, b1, a1, hbuf, N_NODES);

    // --- layer 2: Tx0 = h ---
    hipMemsetAsync(Tx1, 0, (size_t)N_NODES * 128 * 4, stream);
    prop_kernel<<<propBlocks, TPB, 0, stream>>>(hbuf, ei, wE, Tx1, 1.0f, N_EDGES);
    init_neg_kernel<<<featBlocks, TPB, 0, stream>>>(hbuf, Tx2, N_NODES * 128);
    prop_kernel<<<propBlocks, TPB, 0, stream>>>(Tx1, ei, wE, Tx2, 2.0f, N_EDGES);
    cheb_gemm_wmma<HID_C, OUT_C><<<gemmGrd2, gemmBlk, 0, stream>>>(
        hbuf, Tx1, Tx2, W2, b2, a2, nodeOut, N_NODES);

    // --- global mean pool ---
    hipMemsetAsync(sums, 0, (size_t)N_GRAPHS * OUT_C * 4, stream);
    hipMemsetAsync(cnt, 0, (size_t)N_GRAPHS * 4, stream);
    pool_accum_kernel<<<(N_NODES * 16 + TPB - 1) / TPB, TPB, 0, stream>>>(
        nodeOut, batch, sums, cnt, N_NODES);
    pool_final_kernel<<<(N_GRAPHS * OUT_C + TPB - 1) / TPB, TPB, 0, stream>>>(
        sums, cnt, outp);
}